// _SharedVectorQuantizer_17008070492263
// MI455X (gfx1250) — compile-verified
//
#include <hip/hip_runtime.h>

#define K_CODES 2048
#define DIM 64
#define NROWS 131072
#define ROWS_PER_BLOCK 256
#define VQ_BLOCKS (NROWS / ROWS_PER_BLOCK) /* 512 */
#define CHUNK 64       /* codebook entries staged in LDS per iteration */
#define LDS_STRIDE 68  /* 64 data + e2 at [64] + zero pad: conflict-free */

typedef float v2f __attribute__((ext_vector_type(2)));
typedef float v8f __attribute__((ext_vector_type(8)));

__device__ __forceinline__ v8f wmma4(v2f a, v2f b, v8f c) {
  return __builtin_amdgcn_wmma_f32_16x16x4_f32(
      /*neg_a=*/false, a, /*neg_b=*/false, b,
      /*c_mod=*/(short)0, c, /*reuse_a=*/false, /*reuse_b=*/false);
}

// ---------------------------------------------------------------------------
// Prep: en2[k] = ||emb_k||^2
// ---------------------------------------------------------------------------
__global__ __launch_bounds__(256) void vq_prep(const float* __restrict__ emb,
                                               float* __restrict__ en2) {
  int k = blockIdx.x * blockDim.x + threadIdx.x;
  if (k < K_CODES) {
    const float4* r = (const float4*)(emb + (size_t)k * DIM);
    float s = 0.f;
#pragma unroll
    for (int i = 0; i < DIM / 4; ++i) {
      float4 v = r[i];
      s = fmaf(v.x, v.x, s);
      s = fmaf(v.y, v.y, s);
      s = fmaf(v.z, v.z, s);
      s = fmaf(v.w, v.w, s);
    }
    en2[k] = s;
  }
}

// ---------------------------------------------------------------------------
// Main kernel. Per wave: two 16-row A tiles (32 rows), 4 independent WMMA
// accumulator chains, B slabs + e2 staged in LDS (double-buffered via regs).
// ---------------------------------------------------------------------------
__global__ __launch_bounds__(256) void vq_main(const float* __restrict__ z,
                                               const float* __restrict__ emb,
                                               const float* __restrict__ en2,
                                               float* __restrict__ zq_out,
                                               float* __restrict__ codes_out,
                                               float* __restrict__ partials) {
  __shared__ float semb[CHUNK * LDS_STRIDE];
  __shared__ float wls[8];

  const int tid  = threadIdx.x;
  const int wave = tid >> 5;
  const int lane = tid & 31;
  const int hi   = lane >> 4;
  const int lo   = lane & 15;
  const int rowBase = blockIdx.x * ROWS_PER_BLOCK + wave * 32;

  // ---- A matrices (two 16-row tiles), 16x4 f32 slab layout:
  // lane<16 holds (row=lane, K={0,1}); lane>=16 holds (row, K={2,3}).
  const float* zrow0 = z + (size_t)(rowBase + lo) * DIM;
  const float* zrow1 = z + (size_t)(rowBase + 16 + lo) * DIM;
  v2f a0[16], a1[16];
#pragma unroll
  for (int k4 = 0; k4 < 16; ++k4) {
    a0[k4] = *(const v2f*)(zrow0 + 4 * k4 + 2 * hi);
    a1[k4] = *(const v2f*)(zrow1 + 4 * k4 + 2 * hi);
  }

  // ---- Staging map: thread covers entry (tid>>2), 16-float piece (tid&3).
  // Threads with sp==3 additionally stage {en2, 0, 0, 0} into the pad slot.
  const int se = tid >> 2;
  const int sp = tid & 3;
  float2* sdst = (float2*)(semb + se * LDS_STRIDE + sp * 16);
  float4* edst = (float4*)(semb + se * LDS_STRIDE + 64);

  float2 stg[8];
  float  stge2 = 0.f;
#pragma unroll
  for (int i = 0; i < 8; ++i)  // chunk 0 into registers
    stg[i] = ((const float2*)(emb + (size_t)se * DIM + sp * 16))[i];
  if (sp == 3) stge2 = en2[se];

  float best0[8], best1[8];
  int   bidx0[8], bidx1[8];
#pragma unroll
  for (int r = 0; r < 8; ++r) {
    best0[r] = 3.4e38f; best1[r] = 3.4e38f;
    bidx0[r] = 0;       bidx1[r] = 0;
  }

  for (int outer = 0; outer < K_CODES / CHUNK; ++outer) {
    const int entryBase = outer * CHUNK;
    __syncthreads();
#pragma unroll
    for (int i = 0; i < 8; ++i) sdst[i] = stg[i];
    if (sp == 3) {
      float4 e4 = {stge2, 0.f, 0.f, 0.f};
      *edst = e4;
    }
    if (outer + 1 < K_CODES / CHUNK) {  // prefetch next chunk into registers
      const float2* nsrc =
          (const float2*)(emb + (size_t)(entryBase + CHUNK + se) * DIM + sp * 16);
#pragma unroll
      for (int i = 0; i < 8; ++i) stg[i] = nsrc[i];
      if (sp == 3) stge2 = en2[entryBase + CHUNK + se];
    }
    __syncthreads();

#pragma unroll 1
    for (int sub = 0; sub < 4; ++sub) {
      const int nIdx = entryBase + sub * 16 + lo;
      const float* rbase = semb + (sub * 16 + lo) * LDS_STRIDE;
      // B slab 4x16: lane<16 -> (K={0,1}, N=lane); lane>=16 -> (K={2,3}, N=lane-16)
      const float* brow = rbase + 2 * hi;
      v2f b[16];
#pragma unroll
      for (int k4 = 0; k4 < 16; ++k4) b[k4] = *(const v2f*)(brow + 4 * k4);
      const float e2 = rbase[64];  // ds_load_b32, latency hidden by WMMA chain

      v8f c0a = {}, c0b = {}, c1a = {}, c1b = {};
#pragma unroll
      for (int k4 = 0; k4 < 8; ++k4) {   // 4 independent chains, B reused 2x
        c0a = wmma4(a0[k4],     b[k4],     c0a);
        c1a = wmma4(a1[k4],     b[k4],     c1a);
        c0b = wmma4(a0[k4 + 8], b[k4 + 8], c0b);
        c1b = wmma4(a1[k4 + 8], b[k4 + 8], c1b);
      }

#pragma unroll
      for (int r = 0; r < 8; ++r) {  // C layout: VGPR r -> (M = r + 8*hi, N = lo)
        float s0 = fmaf(-2.f, c0a[r], fmaf(-2.f, c0b[r], e2));
        bool  p0 = s0 < best0[r];
        best0[r] = p0 ? s0 : best0[r];
        bidx0[r] = p0 ? nIdx : bidx0[r];

        float s1 = fmaf(-2.f, c1a[r], fmaf(-2.f, c1b[r], e2));
        bool  p1 = s1 < best1[r];
        best1[r] = p1 ? s1 : best1[r];
        bidx1[r] = p1 ? nIdx : bidx1[r];
      }
    }
  }

  // ---- Min-reduce (value,index) across the 16 lanes of each half-wave.
#pragma unroll
  for (int r = 0; r < 8; ++r) {
    float bv0 = best0[r]; int bi0 = bidx0[r];
    float bv1 = best1[r]; int bi1 = bidx1[r];
#pragma unroll
    for (int m = 8; m >= 1; m >>= 1) {
      float ov0 = __shfl_xor(bv0, m, 32);
      int   oi0 = __shfl_xor(bi0, m, 32);
      if (ov0 < bv0 || (ov0 == bv0 && oi0 < bi0)) { bv0 = ov0; bi0 = oi0; }
      float ov1 = __shfl_xor(bv1, m, 32);
      int   oi1 = __shfl_xor(bi1, m, 32);
      if (ov1 < bv1 || (ov1 == bv1 && oi1 < bi1)) { bv1 = ov1; bi1 = oi1; }
    }
    bidx0[r] = bi0;  // all lanes in each half agree: row = r + 8*hi (tile 0)
    bidx1[r] = bi1;  // tile 1
  }

  // ---- Emit z_q rows (gather emb[code]), codes, and loss partial.
  float lsum = 0.f;
#pragma unroll
  for (int m = 0; m < 32; ++m) {
    const int mt = m & 15;
    const int code =
        __shfl((m < 16) ? bidx0[mt & 7] : bidx1[mt & 7], (mt < 8) ? 0 : 16, 32);
    const size_t grow = (size_t)(rowBase + m);
    float2 q  = *(const float2*)(emb + (size_t)code * DIM + 2 * lane);
    float2 zv = *(const float2*)(z + grow * DIM + 2 * lane);
    *(float2*)(zq_out + grow * DIM + 2 * lane) = q;
    float dx = q.x - zv.x, dy = q.y - zv.y;
    lsum = fmaf(dx, dx, lsum);
    lsum = fmaf(dy, dy, lsum);
    if (lane == 0) codes_out[grow] = (float)code;
  }

#pragma unroll
  for (int m = 16; m >= 1; m >>= 1) lsum += __shfl_xor(lsum, m, 32);
  if (lane == 0) wls[wave] = lsum;
  __syncthreads();
  if (tid == 0) {
    float s = 0.f;
#pragma unroll
    for (int w = 0; w < 8; ++w) s += wls[w];
    partials[blockIdx.x] = s;
  }
}

// ---------------------------------------------------------------------------
// Final scalar: commit_loss = 10 * (0.25*L + L) = 12.5 * mean((zq - z)^2)
// ---------------------------------------------------------------------------
__global__ __launch_bounds__(256) void vq_loss(const float* __restrict__ partials,
                                               float* __restrict__ out_loss) {
  __shared__ float red[256];
  float s = 0.f;
  for (int i = threadIdx.x; i < VQ_BLOCKS; i += 256) s += partials[i];
  red[threadIdx.x] = s;
  __syncthreads();
  for (int step = 128; step >= 1; step >>= 1) {
    if (threadIdx.x < step) red[threadIdx.x] += red[threadIdx.x + step];
    __syncthreads();
  }
  if (threadIdx.x == 0) {
    float L = red[0] / (float)((size_t)NROWS * DIM);
    out_loss[0] = 10.0f * (0.25f * L + L);
  }
}

// ---------------------------------------------------------------------------
extern "C" void kernel_launch(void* const* d_in, const int* in_sizes, int n_in,
                              void* d_out, int out_size, void* d_ws, size_t ws_size,
                              hipStream_t stream) {
  const float* z   = (const float*)d_in[0];
  const float* emb = (const float*)d_in[1];

  float* out       = (float*)d_out;
  float* zq_out    = out;                       // [NROWS*DIM]
  float* loss_out  = out + (size_t)NROWS * DIM; // [1]
  float* codes_out = loss_out + 1;              // [NROWS] (as float)

  float* en2      = (float*)d_ws;               // [K_CODES]
  float* partials = en2 + K_CODES;              // [VQ_BLOCKS]

  vq_prep<<<(K_CODES + 255) / 256, 256, 0, stream>>>(emb, en2);
  vq_main<<<VQ_BLOCKS, 256, 0, stream>>>(z, emb, en2, zq_out, codes_out, partials);
  vq_loss<<<1, 256, 0, stream>>>(partials, loss_out);
}